// HollywoodFFTFast_53584011985638
// MI455X (gfx1250) — compile-verified
//
#include <hip/hip_runtime.h>
#include <cstdint>

// ---------------------------------------------------------------------------
// Batched 4096-point complex FFT (radix-2 DIT semantics, computed as 3 radix-16
// passes per row). One workgroup (256 threads, 8 wave32s) per row.
// Input staged into LDS with the Tensor Data Mover (tensor_load_to_lds),
// bit-reversal folded into the pass-0 LDS gather, butterfly storage uses a
// skewed LDS layout (i + (i>>6)) for bank-conflict-free strided access.
// ---------------------------------------------------------------------------

#define FFT_N 4096
#define SKEW_N 4160  // 4096 + 64 skew slack

typedef unsigned int u32;
typedef unsigned long long u64;
typedef __attribute__((ext_vector_type(4))) u32 v4u;
typedef __attribute__((ext_vector_type(8))) int v8i;
typedef __attribute__((ext_vector_type(4))) int v4i;

// Roots of unity exp(-i*pi*j/2^q), q = substage, j = twiddle index.
__device__ constexpr float CRt[4][8] = {
    {1.0f, 0, 0, 0, 0, 0, 0, 0},
    {1.0f, 0.0f, 0, 0, 0, 0, 0, 0},
    {1.0f, 0.70710678118654752f, 0.0f, -0.70710678118654752f, 0, 0, 0, 0},
    {1.0f, 0.92387953251128674f, 0.70710678118654752f, 0.38268343236508977f,
     0.0f, -0.38268343236508977f, -0.70710678118654752f, -0.92387953251128674f}};
__device__ constexpr float CIt[4][8] = {
    {0.0f, 0, 0, 0, 0, 0, 0, 0},
    {0.0f, -1.0f, 0, 0, 0, 0, 0, 0},
    {0.0f, -0.70710678118654752f, -1.0f, -0.70710678118654752f, 0, 0, 0, 0},
    {0.0f, -0.38268343236508977f, -0.70710678118654752f, -0.92387953251128674f,
     -1.0f, -0.92387953251128674f, -0.70710678118654752f, -0.38268343236508977f}};

// 16-point DIT kernel on register-resident data: 4 radix-2 substages covering
// global stages s = 4P..4P+3. L = low 4P bits of the group's global index.
template <int P>
__device__ __forceinline__ void radix16(float (&ar)[16], float (&ai)[16], int L) {
  const float NEG_TWO_PI = -6.283185307179586f;
  float br_[4], bi_[4];
#pragma unroll
  for (int q = 0; q < 4; ++q) {
    // base twiddle angle: -2*pi*L / 2^(4P+q+1)
    float ang = NEG_TWO_PI * (float)L * (1.0f / (float)(1 << (4 * P + q + 1)));
    __sincosf(ang, &bi_[q], &br_[q]);
  }
#pragma unroll
  for (int q = 0; q < 4; ++q) {
    const int stride = 1 << q;
#pragma unroll
    for (int g = 0; g < 8; ++g) {
      const int j = g & (stride - 1);
      const int mu = ((g >> q) << (q + 1)) | j;
      const int lo = mu + stride;
      // w = base_q * exp(-i*pi*j/2^q)
      const float cr = CRt[q][j], ci = CIt[q][j];
      const float wr = br_[q] * cr - bi_[q] * ci;
      const float wi = br_[q] * ci + bi_[q] * cr;
      const float xbr = ar[lo], xbi = ai[lo];
      const float tr = wr * xbr - wi * xbi;
      const float ti = wr * xbi + wi * xbr;
      const float ur = ar[mu], ui = ai[mu];
      ar[mu] = ur + tr;
      ai[mu] = ui + ti;
      ar[lo] = ur - tr;
      ai[lo] = ui - ti;
    }
  }
}

// Issue one TDM 1-D copy: 4096 fp32 from global -> LDS (linear).
__device__ __forceinline__ void tdm_load_row(const float* gptr, float* lptr) {
  u64 ga = (u64)(uintptr_t)gptr;
  u32 lds = (u32)(uintptr_t)lptr;  // the generic LDS pointer's low 32 bits are the LDS byte offset

  v4u g0;
  g0.x = 1u;                                        // count=1, user descriptor
  g0.y = lds;                                       // lds_addr (bytes)
  g0.z = (u32)ga;                                   // global_addr[31:0]
  g0.w = (u32)((ga >> 32) & 0x1FFFFFFu) | (2u << 30);  // global_addr[56:32] | type=2

  v8i g1;
  g1[0] = (int)(2u << 16);        // workgroup_mask=0, data_size=2 (4 bytes)
  g1[1] = (int)(4096u << 16);     // tensor_dim0[15:0]=4096 in bits[63:48]
  g1[2] = (int)(1u << 16);        // tensor_dim0[31:16]=0, tensor_dim1[15:0]=1
  g1[3] = (int)(4096u << 16);     // tensor_dim1[31:16]=0, tile_dim0=4096
  g1[4] = 1;                      // tile_dim1=1, tile_dim2=0
  g1[5] = 4096;                   // tensor_dim0_stride[31:0]
  g1[6] = 0;                      // stride high bits / dim1 stride
  g1[7] = 0;

  v4i z4 = {0, 0, 0, 0};
  v8i z8 = {0, 0, 0, 0, 0, 0, 0, 0};
  // 6-arg form (amdgpu-toolchain / clang-23): groups 2/3 + extra group, cpol.
  __builtin_amdgcn_tensor_load_to_lds(g0, g1, z4, z4, z8, 0);
}

__global__ __launch_bounds__(256) void fft4096_tdm_kernel(
    const float* __restrict__ xr, const float* __restrict__ xi,
    float* __restrict__ yr, float* __restrict__ yi) {
  __shared__ float sRe[SKEW_N];
  __shared__ float sIm[SKEW_N];

  const int t = threadIdx.x;                       // 0..255
  const size_t row = (size_t)blockIdx.x * FFT_N;

  // ---- Stage A: TDM copies this row into the first 4096 floats of sRe/sIm (linear)----------
  if (t < 32) {  // wave 0 issues the DMA for the workgroup
    tdm_load_row(xr + row, &sRe[0]);
    tdm_load_row(xi + row, &sIm[0]);
    __builtin_amdgcn_s_wait_tensorcnt(0);
  }
  __syncthreads();

  float ar[16], ai[16];

  // ---- Pass 0: stages 0..3. Bit-reversal folded into the gather. ----------
  // Thread t handles group G = rev8(t): elements pos = G*16 + mu of the
  // bit-reversed array, i.e. x[rev12(pos)] = LDS element rev4(mu)*256 + t.
  const int rt = (int)(__brev((u32)t) >> 24);  // rev8(t)
#pragma unroll
  for (int mu = 0; mu < 16; ++mu) {
    const int rmu = (int)(__brev((u32)mu) >> 28);  // rev4(mu)
    const int a = rmu * 256 + t;                   // lane-consecutive, no conflict
    ar[mu] = sRe[a];
    ai[mu] = sIm[a];
  }
  radix16<0>(ar, ai, 0);  // L = 0 for the first 4 stages
  __syncthreads();        // all linear reads done before skewed writes clobber
#pragma unroll
  for (int mu = 0; mu < 16; ++mu) {
    const int pos = rt * 16 + mu;
    const int sa = pos + (pos >> 6);  // skew: conflict-free bit-reversed write
    sRe[sa] = ar[mu];
    sIm[sa] = ai[mu];
  }
  __syncthreads();

  // ---- Pass 1: stages 4..7. pos = (t&15)<<8 | mu<<4 | (t>>4), L = t>>4 -----
  const int L1 = t >> 4;
  const int hi1 = t & 15;
#pragma unroll
  for (int mu = 0; mu < 16; ++mu) {
    const int pos = (hi1 << 8) | (mu << 4) | L1;
    const int sa = pos + (pos >> 6);
    ar[mu] = sRe[sa];
    ai[mu] = sIm[sa];
  }
  radix16<1>(ar, ai, L1);
#pragma unroll
  for (int mu = 0; mu < 16; ++mu) {  // in-place within this thread's group
    const int pos = (hi1 << 8) | (mu << 4) | L1;
    const int sa = pos + (pos >> 6);
    sRe[sa] = ar[mu];
    sIm[sa] = ai[mu];
  }
  __syncthreads();

  // ---- Pass 2: stages 8..11. pos = mu<<8 | t, L = t ------------------------
#pragma unroll
  for (int mu = 0; mu < 16; ++mu) {
    const int pos = (mu << 8) | t;
    const int sa = pos + (pos >> 6);
    ar[mu] = sRe[sa];
    ai[mu] = sIm[sa];
  }
  radix16<2>(ar, ai, t);

  // ---- Output: element pos = mu*256 + t -> fully coalesced global stores ---
#pragma unroll
  for (int mu = 0; mu < 16; ++mu) {
    const size_t o = row + (size_t)(mu * 256 + t);
    yr[o] = ar[mu];
    yi[o] = ai[mu];
  }
}

extern "C" void kernel_launch(void* const* d_in, const int* in_sizes, int n_in,
                              void* d_out, int out_size, void* d_ws, size_t ws_size,
                              hipStream_t stream) {
  (void)n_in;
  (void)d_ws;
  (void)ws_size;
  (void)out_size;
  const float* xr = (const float*)d_in[0];
  const float* xi = (const float*)d_in[1];
  const int B = in_sizes[0] / FFT_N;  // 8192 rows
  float* yr = (float*)d_out;
  float* yi = yr + (size_t)B * FFT_N;
  hipLaunchKernelGGL(fft4096_tdm_kernel, dim3(B), dim3(256), 0, stream, xr, xi,
                     yr, yi);
}